// RecurrentOperator_45853070852493
// MI455X (gfx1250) — compile-verified
//
#include <hip/hip_runtime.h>
#include <math.h>

// Problem constants (from reference): B=4, T=8192, D=1024
#define B_    4
#define T_    8192
#define D_    1024
#define N2_   (2 * D_)          // 2048 output gate dims
#define M_    (B_ * T_)         // 32768 flattened rows
#define NCH_  64                // scan chunks along T
#define CHUNK_ (T_ / NCH_)      // 128 timesteps per chunk
#define BD_   (B_ * D_)         // 4096 channels

typedef __attribute__((ext_vector_type(2))) float v2f;
typedef __attribute__((ext_vector_type(8))) float v8f;

// sigmoid via v_exp_f32 + v_rcp_f32 (no IEEE divide expansion)
__device__ __forceinline__ float fast_sigmoid(float g) {
    return __builtin_amdgcn_rcpf(1.0f + __expf(-g));
}

// ---------------------------------------------------------------------------
// Kernel 1: gates = features @ W^T + b, fused sigmoid + split + u = i*x.
// One wave computes one 16x16 tile via V_WMMA_F32_16X16X4_F32 (exact fp32);
// 4 waves/block cover a 16x64 tile. C tiles are staged through LDS so the
// epilogue (bias + sigmoid [+ *x]) reads/writes row-contiguous float4.
// IGATE is a template parameter: the i-half and f-half are separate launches,
// so there is no divergence and no exec-mask juggling.
//
// WMMA 16x16x4 f32 fragment layout (ISA 7.12.2):
//   A (16x4): lane L<16 holds A[M=L, K=k..k+1]; lane L>=16 holds K=k+2..k+3.
//   B (4x16): mirrored; B[k,n] = W[n*D + k] -> aligned float2 loads.
//   C/D: VGPR j, lane L: row M = j + 8*(L>=16), col N = L%16.
// ---------------------------------------------------------------------------
template <bool IGATE>
__global__ __launch_bounds__(128) void gates_gemm_wmma(
    const float* __restrict__ x,
    const float* __restrict__ features,
    const float* __restrict__ W,
    const float* __restrict__ bias,
    float* __restrict__ out_ws)        // u_ws (IGATE) or f_ws
{
    __shared__ __align__(16) float lds[16 * 64];

    const int tid  = threadIdx.x;
    const int lane = tid & 31;
    const int wave = tid >> 5;
    const int ncol0 = blockIdx.x * 64;             // half-space col base [0,1024)
    const int n0    = ncol0 + wave * 16;           // this wave's col tile
    const int gcol  = n0 + (IGATE ? 0 : D_);       // gate-space column for W/bias
    const int m0    = blockIdx.y * 16;
    const int half  = lane >> 4;                   // 0: lanes 0-15, 1: 16-31
    const int l16   = lane & 15;

    const float* __restrict__ ap = features + (size_t)(m0 + l16) * D_;
    const float* __restrict__ bp = W + (size_t)(gcol + l16) * D_;

    v8f acc = {};
#pragma unroll 4
    for (int k = 0; k < D_; k += 4) {
        const int ka = k + 2 * half;               // K pair for this lane half
        v2f a = *(const v2f*)(ap + ka);            // 8B aligned (ka even)
        v2f b = *(const v2f*)(bp + ka);
        acc = __builtin_amdgcn_wmma_f32_16x16x4_f32(
            /*neg_a=*/false, a, /*neg_b=*/false, b,
            /*c_mod=*/(short)0, acc, /*reuse_a=*/false, /*reuse_b=*/false);
    }

    // Stage C tile to LDS as [16 rows][64 cols] (row-contiguous; lanes of a
    // half write 16 consecutive banks -> conflict-free).
#pragma unroll
    for (int j = 0; j < 8; ++j) {
        lds[(j + 8 * half) * 64 + wave * 16 + l16] = acc[j];
    }
    __syncthreads();

    // Epilogue: 128 threads x 8 floats (2x float4), row-contiguous.
    const int row = tid >> 3;                      // 0..15
    const int cq  = (tid & 7) * 8;                 // 0..56
    const size_t gbase = (size_t)(m0 + row) * D_ + ncol0 + cq;  // half-space d
    const float* __restrict__ bptr = bias + (IGATE ? 0 : D_) + ncol0 + cq;

#pragma unroll
    for (int v = 0; v < 2; ++v) {
        const float4 g4 = *(const float4*)(&lds[row * 64 + cq + v * 4]);
        const float4 b4 = *(const float4*)(bptr + v * 4);
        float4 r;
        r.x = fast_sigmoid(g4.x + b4.x);
        r.y = fast_sigmoid(g4.y + b4.y);
        r.z = fast_sigmoid(g4.z + b4.z);
        r.w = fast_sigmoid(g4.w + b4.w);
        if (IGATE) {
            const float4 x4 = *(const float4*)(x + gbase + v * 4);
            r.x *= x4.x; r.y *= x4.y; r.z *= x4.z; r.w *= x4.w;
        }
        *(float4*)(out_ws + gbase + v * 4) = r;
    }
}

// ---------------------------------------------------------------------------
// Kernel 2: per-chunk reduction of the scan monoid (A = prod f, U = local h).
// Each thread handles 4 consecutive channels -> b128 loads, fully coalesced.
// ---------------------------------------------------------------------------
__global__ __launch_bounds__(256) void scan_chunk_reduce(
    const float* __restrict__ f_ws,
    const float* __restrict__ u_ws,
    float* __restrict__ Aagg,
    float* __restrict__ Uagg)
{
    const int d4 = (blockIdx.x * 256 + threadIdx.x) * 4;   // 0..D-4
    const int c  = blockIdx.y;
    const int b  = blockIdx.z;
    const size_t base = ((size_t)b * T_ + (size_t)c * CHUNK_) * D_ + d4;

    float4 a = make_float4(1.f, 1.f, 1.f, 1.f);
    float4 h = make_float4(0.f, 0.f, 0.f, 0.f);
#pragma unroll 4
    for (int t = 0; t < CHUNK_; ++t) {
        const float4 fv = *(const float4*)(f_ws + base + (size_t)t * D_);
        const float4 uv = *(const float4*)(u_ws + base + (size_t)t * D_);
        a.x *= fv.x; a.y *= fv.y; a.z *= fv.z; a.w *= fv.w;
        h.x = fmaf(fv.x, h.x, uv.x);
        h.y = fmaf(fv.y, h.y, uv.y);
        h.z = fmaf(fv.z, h.z, uv.z);
        h.w = fmaf(fv.w, h.w, uv.w);
    }
    const size_t agg = (size_t)c * BD_ + b * D_ + d4;  // [chunk][channel]
    *(float4*)(Aagg + agg) = a;
    *(float4*)(Uagg + agg) = h;
}

// ---------------------------------------------------------------------------
// Kernel 3: sequential scan over the 64 chunk aggregates per channel,
// emitting the carry entering each chunk. ~3 MB total -> L2 resident.
// carry(0)=0; carry(c+1) = A_c*carry(c) + U_c. 4 channels per thread.
// ---------------------------------------------------------------------------
__global__ __launch_bounds__(256) void scan_carry(
    const float* __restrict__ Aagg,
    const float* __restrict__ Uagg,
    float* __restrict__ carry)
{
    const int ch4 = (blockIdx.x * 256 + threadIdx.x) * 4;  // 0..B*D-4
    float4 h = make_float4(0.f, 0.f, 0.f, 0.f);
    for (int c = 0; c < NCH_; ++c) {
        const size_t idx = (size_t)c * BD_ + ch4;
        *(float4*)(carry + idx) = h;
        const float4 a = *(const float4*)(Aagg + idx);
        const float4 u = *(const float4*)(Uagg + idx);
        h.x = fmaf(a.x, h.x, u.x);
        h.y = fmaf(a.y, h.y, u.y);
        h.z = fmaf(a.z, h.z, u.z);
        h.w = fmaf(a.w, h.w, u.w);
    }
}

// ---------------------------------------------------------------------------
// Kernel 4: re-run the recurrence inside each chunk seeded with its carry,
// writing the final h. Same coalesced float4 mapping as kernel 2.
// ---------------------------------------------------------------------------
__global__ __launch_bounds__(256) void scan_apply(
    const float* __restrict__ f_ws,
    const float* __restrict__ u_ws,
    const float* __restrict__ carry,
    float* __restrict__ out)
{
    const int d4 = (blockIdx.x * 256 + threadIdx.x) * 4;
    const int c  = blockIdx.y;
    const int b  = blockIdx.z;
    const size_t base = ((size_t)b * T_ + (size_t)c * CHUNK_) * D_ + d4;

    float4 h = *(const float4*)(carry + (size_t)c * BD_ + b * D_ + d4);
#pragma unroll 4
    for (int t = 0; t < CHUNK_; ++t) {
        const size_t idx = base + (size_t)t * D_;
        const float4 fv = *(const float4*)(f_ws + idx);
        const float4 uv = *(const float4*)(u_ws + idx);
        h.x = fmaf(fv.x, h.x, uv.x);
        h.y = fmaf(fv.y, h.y, uv.y);
        h.z = fmaf(fv.z, h.z, uv.z);
        h.w = fmaf(fv.w, h.w, uv.w);
        *(float4*)(out + idx) = h;
    }
}

// ---------------------------------------------------------------------------
// Workspace layout (floats):
//   f_ws : M_*D_    = 33,554,432  (128 MB)
//   u_ws : M_*D_    = 33,554,432  (128 MB)
//   Aagg : NCH_*BD_ =    262,144  (1 MB)
//   Uagg : NCH_*BD_ =    262,144  (1 MB)
//   carry: NCH_*BD_ =    262,144  (1 MB)
// ---------------------------------------------------------------------------
extern "C" void kernel_launch(void* const* d_in, const int* in_sizes, int n_in,
                              void* d_out, int out_size, void* d_ws, size_t ws_size,
                              hipStream_t stream)
{
    const float* x        = (const float*)d_in[0];
    const float* features = (const float*)d_in[1];
    const float* W        = (const float*)d_in[2];
    const float* bias     = (const float*)d_in[3];
    float* out = (float*)d_out;

    float* ws    = (float*)d_ws;
    float* f_ws  = ws;
    float* u_ws  = f_ws + (size_t)M_ * D_;
    float* Aagg  = u_ws + (size_t)M_ * D_;
    float* Uagg  = Aagg + (size_t)NCH_ * BD_;
    float* carry = Uagg + (size_t)NCH_ * BD_;

    // GEMM halves: 16 n-blocks x 2048 m-tiles each, 4 waves (16x16 tiles)/block.
    dim3 ggrid(D_ / 64, M_ / 16);
    gates_gemm_wmma<true ><<<ggrid, 128, 0, stream>>>(x, features, W, bias, u_ws);
    gates_gemm_wmma<false><<<ggrid, 128, 0, stream>>>(x, features, W, bias, f_ws);

    // Scan phase 1: chunk aggregates (4 channels/thread).
    dim3 sgrid(D_ / 1024, NCH_, B_);
    scan_chunk_reduce<<<sgrid, 256, 0, stream>>>(f_ws, u_ws, Aagg, Uagg);

    // Scan phase 2: carry per (chunk, channel).
    scan_carry<<<BD_ / 1024, 256, 0, stream>>>(Aagg, Uagg, carry);

    // Scan phase 3: apply carries, write h.
    scan_apply<<<sgrid, 256, 0, stream>>>(f_ws, u_ws, carry, out);
}